// SinkhornDistance_45973329936821
// MI455X (gfx1250) — compile-verified
//
#include <hip/hip_runtime.h>
#include <math.h>

#define NPT 2048          // points per side
#define DIM 64            // feature dim
#define EPSV 0.1f
#define INV_EPS 10.0f
#define MAX_ITER 10
#define LOG_MU (-7.62459852f)   // log(1/2048 + 1e-8)
#define LOG_NU (-7.62459852f)

typedef __attribute__((ext_vector_type(2))) float v2f;
typedef __attribute__((ext_vector_type(8))) float v8f;

// workspace layout (in floats)
#define WS_DONE  0                 // int flag at ws[0]
#define WS_XN    16                // xn[2048]
#define WS_YN    (16 + NPT)        // yn[2048]
#define WS_U     (16 + 2*NPT)      // u[2048]
#define WS_V     (16 + 3*NPT)      // v[2048]
#define WS_ERR   (16 + 4*NPT)      // per-row |u_new-u|
#define WS_BSUM  (16 + 5*NPT)      // per-row partial cost sums
#define WS_TOTAL (16 + 6*NPT)

// ---------------- init: zero state each launch (graph-replay safe) -------
__global__ void sink_init(float* __restrict__ ws, float* __restrict__ out) {
  for (int i = threadIdx.x; i < WS_TOTAL; i += blockDim.x) ws[i] = 0.0f;
  if (threadIdx.x == 0) out[0] = 0.0f;
}

// ---------------- row squared-norms of x and y ---------------------------
__global__ void sink_norms(const float* __restrict__ x, const float* __restrict__ y,
                           float* __restrict__ ws) {
  __shared__ float red[64];
  const int row = blockIdx.x;                 // 0..4095
  const int isY = (row >= NPT);
  const int r = isY ? row - NPT : row;
  const float* src = isY ? y : x;
  float v = src[r * DIM + threadIdx.x];
  red[threadIdx.x] = v * v;
  __syncthreads();
  for (int s = 32; s > 0; s >>= 1) {
    if (threadIdx.x < s) red[threadIdx.x] += red[threadIdx.x + s];
    __syncthreads();
  }
  if (threadIdx.x == 0) ws[(isY ? WS_YN : WS_XN) + r] = red[0];
}

// ---------------- C = xn + yn - 2*x.y^T via V_WMMA_F32_16X16X4_F32 -------
// One wave per 16x16 output tile; 8 waves (tiles) per 256-thread block.
// f32 A-layout (16x4): lanes 0-15 hold {K0,K1}, lanes 16-31 hold {K2,K3}.
__global__ void sink_cost_wmma(const float* __restrict__ x, const float* __restrict__ y,
                               const float* __restrict__ ws, float* __restrict__ Cout) {
  const int wave   = threadIdx.x >> 5;
  const int lane   = threadIdx.x & 31;
  const int tileId = blockIdx.x * 8 + wave;   // 0..16383
  const int ti = tileId >> 7;                 // 0..127
  const int tj = tileId & 127;                // 0..127
  const int i0 = ti * 16, j0 = tj * 16;
  const int half = lane >> 4;                 // 0 or 1
  const int m    = lane & 15;

  v8f acc = {0.f,0.f,0.f,0.f,0.f,0.f,0.f,0.f};
#pragma unroll
  for (int k0 = 0; k0 < DIM; k0 += 4) {
    const int ka = k0 + 2 * half;
    v2f a = *(const v2f*)(x + (i0 + m) * DIM + ka);   // A: x tile rows
    v2f b = *(const v2f*)(y + (j0 + m) * DIM + ka);   // B: y tile rows (B = y^T chunk)
    acc = __builtin_amdgcn_wmma_f32_16x16x4_f32(
        /*neg_a=*/false, a, /*neg_b=*/false, b,
        /*c_mod=*/(short)0, acc, /*reuse_a=*/false, /*reuse_b=*/false);
  }

  const float* xn = ws + WS_XN;
  const float ynv = ws[WS_YN + j0 + m];       // N index = lane&15
#pragma unroll
  for (int r = 0; r < 8; ++r) {
    const int gi = i0 + r + 8 * half;         // M index per C/D layout
    Cout[(size_t)gi * NPT + (j0 + m)] = xn[gi] + ynv - 2.0f * acc[r];
  }
}

// ---------------- row pass: u update (logsumexp over j) ------------------
__global__ void sink_row(const float* __restrict__ C, float* __restrict__ ws) {
  __shared__ float red[256];
  const int i = blockIdx.x;
  const int t = threadIdx.x;
  float* u = ws + WS_U;
  const float* v = ws + WS_V;
  const float ui = u[i];
  const float* Crow = C + (size_t)i * NPT;

  float mx = -3.4e38f;
  for (int j = t; j < NPT; j += 256)
    mx = fmaxf(mx, (ui + v[j] - Crow[j]) * INV_EPS);
  red[t] = mx; __syncthreads();
  for (int s = 128; s > 0; s >>= 1) { if (t < s) red[t] = fmaxf(red[t], red[t + s]); __syncthreads(); }
  mx = red[0]; __syncthreads();

  float sum = 0.0f;
  for (int j = t; j < NPT; j += 256)
    sum += __expf((ui + v[j] - Crow[j]) * INV_EPS - mx);
  red[t] = sum; __syncthreads();
  for (int s = 128; s > 0; s >>= 1) { if (t < s) red[t] += red[t + s]; __syncthreads(); }

  if (t == 0) {
    const float lse = mx + __logf(red[0]);
    const float un = EPSV * (LOG_MU - lse) + ui;
    ws[WS_ERR + i] = fabsf(un - ui);
    const int done = ((const int*)ws)[WS_DONE];
    u[i] = done ? ui : un;
  }
}

// ---------------- col pass: v update (logsumexp over i) ------------------
// 64-column slab per block; lanes map to consecutive columns (coalesced).
__global__ void sink_col(const float* __restrict__ C, float* __restrict__ ws) {
  __shared__ float red[256];
  const int c  = threadIdx.x & 63;
  const int rg = threadIdx.x >> 6;            // 0..3 row groups
  const int j  = blockIdx.x * 64 + c;
  const float* u = ws + WS_U;
  float* v = ws + WS_V;
  const float vj = v[j];

  float mx = -3.4e38f;
  for (int i = rg; i < NPT; i += 4)
    mx = fmaxf(mx, (u[i] + vj - C[(size_t)i * NPT + j]) * INV_EPS);
  red[threadIdx.x] = mx; __syncthreads();
  mx = fmaxf(fmaxf(red[c], red[c + 64]), fmaxf(red[c + 128], red[c + 192]));
  __syncthreads();

  float sum = 0.0f;
  for (int i = rg; i < NPT; i += 4)
    sum += __expf((u[i] + vj - C[(size_t)i * NPT + j]) * INV_EPS - mx);
  red[threadIdx.x] = sum; __syncthreads();

  if (rg == 0) {
    const float tot = red[c] + red[c + 64] + red[c + 128] + red[c + 192];
    const float lse = mx + __logf(tot);
    const float vn = EPSV * (LOG_NU - lse) + vj;
    const int done = ((const int*)ws)[WS_DONE];
    v[j] = done ? vj : vn;
  }
}

// ---------------- done-flag update: deterministic err reduction ----------
__global__ void sink_done(float* __restrict__ ws) {
  __shared__ float red[256];
  const int t = threadIdx.x;
  float s = 0.0f;
  for (int i = t; i < NPT; i += 256) s += ws[WS_ERR + i];
  red[t] = s; __syncthreads();
  for (int k = 128; k > 0; k >>= 1) { if (t < k) red[t] += red[t + k]; __syncthreads(); }
  if (t == 0 && red[0] < 0.1f) ((int*)ws)[WS_DONE] = 1;
}

// ---------------- pi = exp(M), per-row partial cost ----------------------
__global__ void sink_pi(const float* __restrict__ C, const float* __restrict__ ws,
                        float* __restrict__ pi, float* __restrict__ bsum) {
  __shared__ float red[256];
  const int i = blockIdx.x;
  const int t = threadIdx.x;
  const float ui = ws[WS_U + i];
  const float* v = ws + WS_V;
  const float* Crow = C + (size_t)i * NPT;
  float* prow = pi + (size_t)i * NPT;
  float local = 0.0f;
  for (int j = t; j < NPT; j += 256) {
    const float cv = Crow[j];
    const float p = __expf((ui + v[j] - cv) * INV_EPS);
    prow[j] = p;
    local += p * cv;
  }
  red[t] = local; __syncthreads();
  for (int s = 128; s > 0; s >>= 1) { if (t < s) red[t] += red[t + s]; __syncthreads(); }
  if (t == 0) bsum[i] = red[0];
}

// ---------------- deterministic final cost sum ---------------------------
__global__ void sink_final(const float* __restrict__ bsum, float* __restrict__ out) {
  __shared__ float red[256];
  const int t = threadIdx.x;
  float s = 0.0f;
  for (int i = t; i < NPT; i += 256) s += bsum[i];
  red[t] = s; __syncthreads();
  for (int k = 128; k > 0; k >>= 1) { if (t < k) red[t] += red[t + k]; __syncthreads(); }
  if (t == 0) out[0] = red[0];
}

extern "C" void kernel_launch(void* const* d_in, const int* in_sizes, int n_in,
                              void* d_out, int out_size, void* d_ws, size_t ws_size,
                              hipStream_t stream) {
  const float* x = (const float*)d_in[0];
  const float* y = (const float*)d_in[1];
  float* out  = (float*)d_out;
  float* pi   = out + 1;                       // pi[2048*2048]
  float* Cmat = out + 1 + (size_t)NPT * NPT;   // C[2048*2048]
  float* ws   = (float*)d_ws;

  sink_init <<<1, 256, 0, stream>>>(ws, out);
  sink_norms<<<2 * NPT, 64, 0, stream>>>(x, y, ws);
  sink_cost_wmma<<<(NPT / 16) * (NPT / 16) / 8, 256, 0, stream>>>(x, y, ws, Cmat);
  for (int it = 0; it < MAX_ITER; ++it) {
    sink_row <<<NPT, 256, 0, stream>>>(Cmat, ws);
    sink_col <<<NPT / 64, 256, 0, stream>>>(Cmat, ws);
    sink_done<<<1, 256, 0, stream>>>(ws);
  }
  sink_pi   <<<NPT, 256, 0, stream>>>(Cmat, ws, pi, ws + WS_BSUM);
  sink_final<<<1, 256, 0, stream>>>(ws + WS_BSUM, out);
}